// GMConv_51565377355978
// MI455X (gfx1250) — compile-verified
//
#include <hip/hip_runtime.h>

// Problem constants (from reference)
#define B_   32
#define CIN  64
#define O_   128
#define K_   9
#define KP   12      // K padded to multiple of 4 for 16x16x4 f32 WMMA
#define L_   4096

typedef __attribute__((ext_vector_type(2))) float v2f;
typedef __attribute__((ext_vector_type(8))) float v8f;

#if __has_builtin(__builtin_amdgcn_wmma_f32_16x16x4_f32)
#define HAVE_WMMA_F32X4 1
#else
#define HAVE_WMMA_F32X4 0
#endif

// ---------------------------------------------------------------------------
// Kernel 1: normalize err_vector -> evn[L]
// ---------------------------------------------------------------------------
__global__ void __launch_bounds__(256) prep_ev(const float* __restrict__ ev,
                                               float* __restrict__ evn) {
    __shared__ float red[256];
    float s = 0.f;
    for (int l = threadIdx.x; l < L_; l += 256) { float v = ev[l]; s += v * v; }
    red[threadIdx.x] = s;
    __syncthreads();
    for (int st = 128; st > 0; st >>= 1) {
        if ((int)threadIdx.x < st) red[threadIdx.x] += red[threadIdx.x + st];
        __syncthreads();
    }
    float r = rsqrtf(red[0]);
    for (int l = threadIdx.x; l < L_; l += 256) evn[l] = ev[l] * r;
}

// ---------------------------------------------------------------------------
// Kernel 2: normalize weight_coeff rows, store K-padded (zeros at k=9..11)
// wcn layout: [O][KP]
// ---------------------------------------------------------------------------
__global__ void __launch_bounds__(128) prep_wc(const float* __restrict__ wc,
                                               float* __restrict__ wcn) {
    int o = threadIdx.x;  // 128 threads, 1 block
    float v[K_];
    float s = 0.f;
#pragma unroll
    for (int k = 0; k < K_; ++k) { v[k] = wc[o * K_ + k]; s += v[k] * v[k]; }
    float r = rsqrtf(s);
#pragma unroll
    for (int k = 0; k < K_; ++k) wcn[o * KP + k] = v[k] * r;
#pragma unroll
    for (int k = K_; k < KP; ++k) wcn[o * KP + k] = 0.f;
}

// ---------------------------------------------------------------------------
// Kernel 3: xsum[b,l] = sum_i x[b,i,0,l]   (coalesced in l)
// ---------------------------------------------------------------------------
__global__ void __launch_bounds__(256) xsum_k(const float* __restrict__ x,
                                              float* __restrict__ xs) {
    int t = blockIdx.x * 256 + threadIdx.x;   // t < B_*L_ (exact)
    int b = t >> 12;                           // / L_
    int l = t & (L_ - 1);
    const float* p = x + (size_t)b * CIN * L_ + l;
    float s = 0.f;
#pragma unroll
    for (int i = 0; i < CIN; ++i) s += p[(size_t)i * L_];
    xs[t] = s;
}

// ---------------------------------------------------------------------------
// Kernel 4: sbuf[b,k,l] = xsum[b, idx[k,l]] * evn[l]  (k<9), zero pad k=9..11
// sbuf layout: [B][KP][L]  (6.3 MB -> L2 resident for the GEMM)
// ---------------------------------------------------------------------------
__global__ void __launch_bounds__(256) gather_k(const float* __restrict__ xs,
                                                const float* __restrict__ evn,
                                                const int* __restrict__ idx,
                                                float* __restrict__ sbuf) {
    int t = blockIdx.x * 256 + threadIdx.x;   // t < B_*KP*L_ (exact)
    int l  = t & (L_ - 1);
    int bk = t >> 12;
    int k  = bk % KP;
    int b  = bk / KP;
    float v = 0.f;
    if (k < K_) v = xs[b * L_ + idx[k * L_ + l]] * evn[l];
    sbuf[t] = v;
}

// ---------------------------------------------------------------------------
// Kernel 5: per-batch GEMM  OUT[o,l] = wcn[o,:] . sbuf[b,:,l] + bias[o]
// via V_WMMA_F32_16X16X4_F32, 3 WMMAs per 16x16 tile (K=12).
// One wave per (b, o-tile, l-group of 8 tiles): 8192 waves.
//
// f32 WMMA fragment layout (ISA 7.12.2):
//   A (16x4): vgpr0 = K{0|2}, vgpr1 = K{1|3}; K+2 on lanes 16..31; M = lane&15
//   B (4x16): symmetric; N = lane&15
//   C/D v8f : vgpr r -> M = r + 8*(lane>=16), N = lane&15
// ---------------------------------------------------------------------------
__global__ void __launch_bounds__(256) wmma_gemm(const float* __restrict__ wcn,
                                                 const float* __restrict__ sbuf,
                                                 const float* __restrict__ bias,
                                                 float* __restrict__ out) {
    int lane = threadIdx.x & 31;
    int w    = blockIdx.x * 8 + (threadIdx.x >> 5);  // 0..8191
    int b    = w >> 8;                                // / 256
    int rem  = w & 255;
    int ot   = rem >> 5;                              // o-tile 0..7
    int ltg  = rem & 31;                              // l-group 0..31
    int ln   = lane & 15;
    int half = lane >> 4;                             // 0 | 1
    int kofs = half * 2;
    int obase = ot * 16;

    // Bias value per accumulator row (uniform within half-wave; L0-cached)
    float bv[8];
#pragma unroll
    for (int r = 0; r < 8; ++r) bv[r] = bias[obase + half * 8 + r];

#if HAVE_WMMA_F32X4
    // A fragments for kk = 0,4,8 (loaded once per wave; wcn table is 6 KB)
    v2f a0, a1, a2;
    {
        const float* wp = wcn + (obase + ln) * KP;
        a0.x = wp[0 + kofs];  a0.y = wp[1 + kofs];
        a1.x = wp[4 + kofs];  a1.y = wp[5 + kofs];
        a2.x = wp[8 + kofs];  a2.y = wp[9 + kofs];
    }
#endif

    const float* sb = sbuf + (size_t)b * KP * L_;
    float*       ob = out  + (size_t)b * O_ * L_;

    for (int it = 0; it < 8; ++it) {
        int l0 = (ltg * 8 + it) * 16;
#if HAVE_WMMA_F32X4
        const float* sp = sb + l0 + ln;
        v2f b0, b1, b2;
        b0.x = sp[(0 + kofs) * L_];  b0.y = sp[(1 + kofs) * L_];
        b1.x = sp[(4 + kofs) * L_];  b1.y = sp[(5 + kofs) * L_];
        b2.x = sp[(8 + kofs) * L_];  b2.y = sp[(9 + kofs) * L_];
        v8f c = {};
        c = __builtin_amdgcn_wmma_f32_16x16x4_f32(false, a0, false, b0,
                                                  (short)0, c, false, false);
        c = __builtin_amdgcn_wmma_f32_16x16x4_f32(false, a1, false, b1,
                                                  (short)0, c, false, false);
        c = __builtin_amdgcn_wmma_f32_16x16x4_f32(false, a2, false, b2,
                                                  (short)0, c, false, false);
#pragma unroll
        for (int r = 0; r < 8; ++r)
            ob[(size_t)(obase + half * 8 + r) * L_ + l0 + ln] = c[r] + bv[r];
#else
        // Fallback: per-lane dot products (no WMMA available)
        float sv[KP];
#pragma unroll
        for (int k = 0; k < KP; ++k) sv[k] = sb[k * L_ + l0 + ln];
#pragma unroll
        for (int r = 0; r < 8; ++r) {
            int o = obase + half * 8 + r;
            float acc = bv[r];
#pragma unroll
            for (int k = 0; k < K_; ++k) acc = fmaf(wcn[o * KP + k], sv[k], acc);
            ob[(size_t)o * L_ + l0 + ln] = acc;
        }
#endif
    }
}

// ---------------------------------------------------------------------------
// Launch
//   d_in: 0=x [B,CIN,1,L] f32, 1=weight_coeff [O,1,K] f32, 2=err_vector [1,L]
//         f32, 3=bias [O] f32, 4=index_matrix [K,L] i32
//   d_out: [B,O,1,L] f32
//   ws (floats): evn@0 (4096) | wcn@4096 (1536) | xsum@5632 (131072)
//                | sbuf@136704 (1572864)   => 6.84 MB total
// ---------------------------------------------------------------------------
extern "C" void kernel_launch(void* const* d_in, const int* in_sizes, int n_in,
                              void* d_out, int out_size, void* d_ws, size_t ws_size,
                              hipStream_t stream) {
    const float* x    = (const float*)d_in[0];
    const float* wc   = (const float*)d_in[1];
    const float* ev   = (const float*)d_in[2];
    const float* bias = (const float*)d_in[3];
    const int*   idx  = (const int*)d_in[4];
    float* out = (float*)d_out;

    float* ws   = (float*)d_ws;
    float* evn  = ws;                       // 4096
    float* wcn  = ws + 4096;                // 1536
    float* xs   = ws + 4096 + 1536;         // 131072
    float* sbuf = ws + 4096 + 1536 + 131072;// B_*KP*L_ = 1572864

    prep_ev <<<1,    256, 0, stream>>>(ev, evn);
    prep_wc <<<1,    128, 0, stream>>>(wc, wcn);
    xsum_k  <<<512,  256, 0, stream>>>(x, xs);          // 512*256 = B*L
    gather_k<<<6144, 256, 0, stream>>>(xs, evn, idx, sbuf); // = B*KP*L
    wmma_gemm<<<1024,256, 0, stream>>>(wcn, sbuf, bias, out); // 8192 waves
}